// PyTorchCustomMappingLayer_9302899163496
// MI455X (gfx1250) — compile-verified
//
#include <hip/hip_runtime.h>

#define BLK 256
#define KSTEPS 10
#define ROW_BYTES 160        // 40 f32 per element
#define IN_STRIDE 176        // padded LDS row stride (44 dwords, 16B aligned)
#define OUT_ROW_BYTES 240    // 60 f32 per element
#define IN_TILE_BYTES (BLK * ROW_BYTES)      // 40960
#define OUT_TILE_BYTES (BLK * OUT_ROW_BYTES) // 61440

// ---- CDNA5 async global<->LDS helpers (gfx1250) ----
// GVS addressing: mem_addr = SGPR64 base + VGPR32 offset.
// LDS address operand is the wave-relative LDS byte offset (low 32 bits of the
// generic pointer to a __shared__ object).
__device__ __forceinline__ unsigned lds_off(const void* p) {
  return (unsigned)(unsigned long long)p;
}
__device__ __forceinline__ void async_load_b128(unsigned lds, unsigned voff,
                                                unsigned long long sbase) {
  asm volatile("global_load_async_to_lds_b128 %0, %1, %2"
               :: "v"(lds), "v"(voff), "s"(sbase) : "memory");
}
__device__ __forceinline__ void async_store_b128(unsigned voff, unsigned lds,
                                                 unsigned long long sbase) {
  asm volatile("global_store_async_from_lds_b128 %0, %1, %2"
               :: "v"(voff), "v"(lds), "s"(sbase) : "memory");
}
__device__ __forceinline__ void wait_async0() {
  asm volatile("s_wait_asynccnt 0" ::: "memory");
}

__global__ __launch_bounds__(BLK) void osu_map_scan_kernel(
    const float* __restrict__ var,        // [B, 40]
    const float* __restrict__ rel,        // [6, 10]
    const float* __restrict__ tickd,      // [10]
    const float* __restrict__ startp,     // [2]
    const float* __restrict__ lenmul,     // [1]
    const unsigned char* __restrict__ nfse_p, // [1] bool
    const float* __restrict__ maxticks,   // [1]
    float* __restrict__ out,              // [B, 10, 6]
    int n)
{
  __shared__ __align__(16) unsigned char smem[OUT_TILE_BYTES]; // reused in/out

  const unsigned t = threadIdx.x;
  const unsigned long long blockElem = (unsigned long long)blockIdx.x * BLK;

  // ---------- stage input tile: global -> LDS via async copies ----------
  const unsigned long long inBase =
      (unsigned long long)var + blockElem * ROW_BYTES;
  const unsigned long long totalIn = (unsigned long long)n * ROW_BYTES;
  const unsigned long long remain = totalIn - blockElem * ROW_BYTES;
  const unsigned tileValid =
      remain >= (unsigned long long)IN_TILE_BYTES ? (unsigned)IN_TILE_BYTES
                                                  : (unsigned)remain;
  const unsigned maxoff = tileValid - 16u;        // clamp target for tail block
  const unsigned smem_base = lds_off(smem);

#pragma unroll
  for (unsigned k = 0; k < 10; ++k) {
    unsigned q = k * BLK + t;                     // 16B-chunk index in tile
    unsigned g = q * 16u;                         // global byte offset
    if (g > maxoff) g = maxoff;                   // OOB-safe (tail block)
    // float idx = q*4 -> element q/10, word (q%10)*4 ; repack to padded rows
    unsigned l = smem_base + (q / 10u) * IN_STRIDE + (q % 10u) * 16u;
    async_load_b128(l, g, inBase);
  }
  wait_async0();
  __syncthreads();

  // ---------- pull my row (40 f32) from LDS into registers ----------
  float r[40];
#pragma unroll
  for (int k = 0; k < 10; ++k) {
    float4 v4 = *reinterpret_cast<const float4*>(smem + t * IN_STRIDE + k * 16);
    r[4 * k + 0] = v4.x; r[4 * k + 1] = v4.y;
    r[4 * k + 2] = v4.z; r[4 * k + 3] = v4.w;
  }
  __syncthreads();   // all rows consumed before LDS is reused for output

  // ---------- uniform scalar parameters ----------
  const float lm = lenmul[0];
  const float mt = maxticks[0];
  const bool  nfse = nfse_p[0] != 0;
  float px = startp[0];
  float py = startp[1];

  // ---------- 10-step scan, fully unrolled ----------
#pragma unroll
  for (int s = 0; s < KSTEPS; ++s) {
    const float is_sl_f = rel[0 * KSTEPS + s];
    const float sln     = rel[1 * KSTEPS + s];
    const float scos    = rel[3 * KSTEPS + s];
    const float ssin    = rel[4 * KSTEPS + s];
    const float ndist   = rel[5 * KSTEPS + s];
    const float l  = lm * ndist;
    const float wl = 0.05f * 512.0f + l * 0.5f;
    const float wr = 0.95f * 512.0f - l * 0.5f;
    const float wt = 0.05f * 384.0f + l * 0.5f;
    const float wb = 0.95f * 384.0f - l * 0.5f;
    const float nr = (tickd[s] <= mt) ? 1.0f : 0.0f;   // use_ds
    const float rr = 1.0f - nr;                        // rerand
    const bool  is_sl = is_sl_f > 0.5f;

    const float vx = r[s];
    const float u  = r[KSTEPS + s];
    const float vy = r[2 * KSTEPS + s];
    const float w  = r[3 * KSTEPS + s];

    const float rx = 256.0f + 256.0f * vx;
    const float ry = 192.0f + 192.0f * vy;
    const float n1 = sqrtf(vx * vx + vy * vy);
    const float c1 = vx / n1, s1 = vy / n1;
    const float n2 = sqrtf(u * u + w * w);
    const float c2 = u / n2,  s2 = w / n2;

    const float dx = l * c1, dy = l * s1;
    const float wvl  = (px < wl) ? 1.0f : 0.0f;
    const float wvr  = (px > wr) ? 1.0f : 0.0f;
    const float wvxm = ((px > wl) && (px < wr)) ? 1.0f : 0.0f;
    const float wvt  = (py < wt) ? 1.0f : 0.0f;
    const float wvb  = (py > wb) ? 1.0f : 0.0f;
    const float wvym = ((py > wt) && (py < wb)) ? 1.0f : 0.0f;
    const float adx = fabsf(dx), ady = fabsf(dy);
    const float xd = adx * wvl - adx * wvr + dx * wvxm;
    const float yd = ady * wvt - ady * wvb + dy * wvym;
    const float x = rr * rx + nr * (px + xd);
    const float y = rr * ry + nr * (py + yd);

    const float oa = c2 * scos - s2 * ssin;
    const float ob = c2 * ssin + s2 * scos;
    const float xe = x + c2 * sln;
    const float ye = y + s2 * sln;
    const float a2 = rr * c2 + nr * c1;
    const float b2 = rr * s2 + nr * s1;

    const float xn = x / 512.0f;
    const float yn = y / 384.0f;

    float* orow = reinterpret_cast<float*>(smem + t * OUT_ROW_BYTES + s * 24);
    orow[0] = xn;
    orow[1] = yn;
    orow[2] = is_sl ? oa : a2;
    orow[3] = is_sl ? ob : b2;
    orow[4] = is_sl ? (xe / 512.0f) : xn;
    orow[5] = is_sl ? (ye / 384.0f) : yn;

    const float px_sl = nfse ? xe : x;
    const float py_sl = nfse ? ye : y;
    px = is_sl ? px_sl : x;
    py = is_sl ? py_sl : y;
  }
  __syncthreads();

  // ---------- output tile: LDS -> global ----------
  const unsigned long long outBase =
      (unsigned long long)out + blockElem * OUT_ROW_BYTES;
  const bool full = (blockElem + BLK) <= (unsigned long long)n;
  if (full) {
#pragma unroll
    for (unsigned c = 0; c < 15; ++c) {
      unsigned off = (c * BLK + t) * 16u;     // linear, coalesced
      async_store_b128(off, smem_base + off, outBase);
    }
    wait_async0();
  } else {
    const unsigned long long i = blockElem + t;
    if (i < (unsigned long long)n) {
      float4* dst = reinterpret_cast<float4*>(out + i * 60);
#pragma unroll
      for (int c = 0; c < 15; ++c) {
        dst[c] = *reinterpret_cast<const float4*>(smem + t * OUT_ROW_BYTES + c * 16);
      }
    }
  }
}

extern "C" void kernel_launch(void* const* d_in, const int* in_sizes, int n_in,
                              void* d_out, int out_size, void* d_ws, size_t ws_size,
                              hipStream_t stream) {
  (void)n_in; (void)out_size; (void)d_ws; (void)ws_size;
  const float* var  = (const float*)d_in[0];   // [B, 40]
  const float* rel  = (const float*)d_in[1];   // [6, 10]
  const float* tick = (const float*)d_in[2];   // [10]
  const float* sp   = (const float*)d_in[3];   // [2]
  const float* lmul = (const float*)d_in[4];   // [1]
  const unsigned char* nfse = (const unsigned char*)d_in[5]; // [1] bool
  const float* mtk  = (const float*)d_in[6];   // [1]
  float* out = (float*)d_out;                  // [B, 10, 6]

  const int B = in_sizes[0] / 40;
  const int grid = (B + BLK - 1) / BLK;
  osu_map_scan_kernel<<<grid, BLK, 0, stream>>>(var, rel, tick, sp, lmul, nfse,
                                                mtk, out, B);
}